// KoLeoLoss_79499844649268
// MI455X (gfx1250) — compile-verified
//
#include <hip/hip_runtime.h>
#include <hip/hip_bf16.h>
#include <math.h>

// Problem constants (reference shapes are fixed: X is 1024 x 256 fp32).
#define KN 1024
#define KD 256
#define KEPS 1e-8f

typedef float v2f __attribute__((ext_vector_type(2)));
typedef float v8f __attribute__((ext_vector_type(8)));

// ---------------------------------------------------------------------------
// Kernel A: row L2 norms via V_WMMA_F32_16X16X4_F32.
// One wave (32 threads) computes 16 row norms: A = 16x4 chunk of X elementwise
// squared, B = ones(4x16), accumulate over K=256 in steps of 4.
// D[m][0] = sum_k A[m][k] -> row sum of squares.
// A-matrix 16x4 f32 layout (ISA 7.12.2): lanes 0-15 hold M=0..15 with K=0 in
// v0, K=1 in v1; lanes 16-31 hold K=2 in v0, K=3 in v1.
// D layout: VGPR r, lanes 0-15 -> (M=r, N=lane), lanes 16-31 -> (M=8+r).
// ---------------------------------------------------------------------------
__global__ void koleo_norms_wmma(const float* __restrict__ X,
                                 float* __restrict__ norms) {
    const int lane   = threadIdx.x;          // 0..31, full wave, EXEC all ones
    const int m      = lane & 15;            // matrix row within tile
    const int khalf  = (lane >> 4) * 2;      // 0 for lanes 0-15, 2 for 16-31
    const int rowbase = blockIdx.x * 16;

    const float* rowp = X + (size_t)(rowbase + m) * KD;

    v8f c = {};
    v2f b = {1.0f, 1.0f};
    for (int k0 = 0; k0 < KD; k0 += 4) {
        float a0 = rowp[k0 + khalf];
        float a1 = rowp[k0 + khalf + 1];
        v2f a = {a0 * a0, a1 * a1};
        // 8 args: (neg_a, A, neg_b, B, c_mod, C, reuse_a, reuse_b)
        c = __builtin_amdgcn_wmma_f32_16x16x4_f32(
            false, a, false, b, (short)0, c, false, false);
    }

    // Column N=0 of D: lane 0 has M=0..7 in c[0..7], lane 16 has M=8..15.
    if ((lane & 15) == 0) {
        const int mbase = rowbase + (lane >> 4) * 8;
#pragma unroll
        for (int r = 0; r < 8; ++r) {
            norms[mbase + r] = sqrtf(c[r]);
        }
    }
}

// ---------------------------------------------------------------------------
// Kernel B: per-column nearest-neighbour with the faithful stride-1025 bug.
// One block per feature k; column staged in LDS (normalized on the fly);
// each of the 256 threads owns 4 rows and scans all 1024 j's.
// Excluded j per (i,k): 769*i + 256*j + k == 0 (mod 1025)
//   -> j_ex = 1021 * ((1025 - (769*i + k) % 1025) % 1025) % 1025
//   (inv(256) mod 1025 == 1021). j_ex == 1024 means nothing excluded.
// ---------------------------------------------------------------------------
__global__ void koleo_main(const float* __restrict__ X,
                           const float* __restrict__ norms,
                           float* __restrict__ partial) {
    __shared__ float col[KN];
    __shared__ float red[256];

    const int k = blockIdx.x;   // feature index 0..255
    const int t = threadIdx.x;  // 0..255

    // Stage normalized column k into LDS (X is L2-resident; strided reads OK).
#pragma unroll
    for (int q = 0; q < 4; ++q) {
        const int r = t + 256 * q;
        const float nv = fmaxf(norms[r], KEPS);
        col[r] = X[(size_t)r * KD + k] / nv;
    }
    __syncthreads();

    float vi[4];
    float mn[4];
    int   jex[4];
#pragma unroll
    for (int q = 0; q < 4; ++q) {
        const int i = t + 256 * q;
        vi[q] = col[i];
        mn[q] = __builtin_inff();
        const int rmod = (769 * i + k) % 1025;
        jex[q] = (int)((1021LL * ((1025 - rmod) % 1025)) % 1025);
    }

    for (int j = 0; j < KN; ++j) {
        const float w = col[j];  // uniform address -> LDS broadcast
#pragma unroll
        for (int q = 0; q < 4; ++q) {
            float dl = fabsf(w - vi[q]);
            dl = (j == jex[q]) ? __builtin_inff() : dl;
            mn[q] = fminf(mn[q], dl);
        }
    }

    float logsum = 0.0f;
#pragma unroll
    for (int q = 0; q < 4; ++q) {
        logsum += logf(mn[q] + KEPS);
    }

    // Fixed-order block tree reduction (deterministic).
    red[t] = logsum;
    __syncthreads();
    for (int s = 128; s > 0; s >>= 1) {
        if (t < s) red[t] += red[t + s];
        __syncthreads();
    }
    if (t == 0) partial[k] = red[0];
}

// ---------------------------------------------------------------------------
// Kernel C: final fixed-order reduction over 256 column partials.
// ---------------------------------------------------------------------------
__global__ void koleo_finish(const float* __restrict__ partial,
                             float* __restrict__ out) {
    __shared__ float red[256];
    const int t = threadIdx.x;
    red[t] = partial[t];
    __syncthreads();
    for (int s = 128; s > 0; s >>= 1) {
        if (t < s) red[t] += red[t + s];
        __syncthreads();
    }
    if (t == 0) out[0] = -red[0] / (float)KN;
}

extern "C" void kernel_launch(void* const* d_in, const int* in_sizes, int n_in,
                              void* d_out, int out_size, void* d_ws, size_t ws_size,
                              hipStream_t stream) {
    (void)in_sizes; (void)n_in; (void)out_size; (void)ws_size;

    const float* X = (const float*)d_in[0];
    float* out = (float*)d_out;

    // Workspace layout: [0,1024) row norms, [1024,1280) column partial sums.
    float* norms   = (float*)d_ws;
    float* partial = norms + KN;

    // A: 64 waves, each computes 16 row norms via WMMA.
    koleo_norms_wmma<<<KN / 16, 32, 0, stream>>>(X, norms);
    // B: one block per feature column.
    koleo_main<<<KD, 256, 0, stream>>>(X, norms, partial);
    // C: scalar result.
    koleo_finish<<<1, 256, 0, stream>>>(partial, out);
}